// SADPixelwise_13838384628330
// MI455X (gfx1250) — compile-verified
//
#include <hip/hip_runtime.h>
#include <math.h>

// SAD pixelwise: sum over N of arccos( dot(in_row, mat[idx]) / (|in_row| * |mat[idx]|) )
// in_row = 31 consecutive floats of the raw-reshaped [B,31,H,W] tensor (faithful to ref).

#define BANDS 31
#define NMAT 16
#define ROWS_PER_BLOCK 128   // 8 waves * 16 rows
#define THREADS 256
#define FLOATS_PER_BLOCK (ROWS_PER_BLOCK * BANDS)   // 3968

typedef float v2f __attribute__((ext_vector_type(2)));
typedef float v8f __attribute__((ext_vector_type(8)));

__device__ __forceinline__ float pick8(v8f v, int c) {
    float r = v[0];
    r = (c == 1) ? v[1] : r;
    r = (c == 2) ? v[2] : r;
    r = (c == 3) ? v[3] : r;
    r = (c == 4) ? v[4] : r;
    r = (c == 5) ? v[5] : r;
    r = (c == 6) ? v[6] : r;
    r = (c == 7) ? v[7] : r;
    return r;
}

#if __has_builtin(__builtin_amdgcn_wmma_f32_16x16x4_f32)
#define HAVE_WMMA_F32 1
__device__ __forceinline__ v8f wmma_f32_k4(v2f a, v2f b, v8f c) {
    // 8 args: (neg_a, A, neg_b, B, c_mod, C, reuse_a, reuse_b)
    return __builtin_amdgcn_wmma_f32_16x16x4_f32(false, a, false, b, (short)0, c,
                                                 false, false);
}
#else
#define HAVE_WMMA_F32 0
#endif

__global__ void __launch_bounds__(THREADS)
sad_partial_kernel(const float* __restrict__ in,
                   const long long* __restrict__ tgt,
                   const float* __restrict__ mat,
                   float* __restrict__ partial,
                   long long Nrows) {
    __shared__ __align__(16) float lds_in[FLOATS_PER_BLOCK];
    __shared__ float lds_tab[17 * BANDS];
    __shared__ float lds_nrm[17];
    __shared__ float lds_part[THREADS / 32];

    const int tid = threadIdx.x;

    // --- Stage material table (+zero row 16) and per-material norms ---
    for (int i = tid; i < 17 * BANDS; i += THREADS) {
        int m = i / BANDS;
        lds_tab[i] = (m < NMAT) ? mat[i] : 0.0f;
    }
    if (tid < 17) {
        float s = 0.0f;
        if (tid < NMAT) {
            #pragma unroll
            for (int c = 0; c < BANDS; ++c) {
                float v = mat[tid * BANDS + c];
                s += v * v;
            }
        }
        lds_nrm[tid] = sqrtf(s);
    }

    // --- Stage 128 input rows with coalesced float4 loads ---
    const long long rowBase = (long long)blockIdx.x * ROWS_PER_BLOCK;
    const long long gOff = rowBase * BANDS;
    long long remain = Nrows * BANDS - gOff;
    int cnt = (remain >= FLOATS_PER_BLOCK) ? FLOATS_PER_BLOCK
                                           : (remain > 0 ? (int)remain : 0);
    {
        const int cnt4 = cnt & ~3;
        for (int f = 4 * tid; f + 4 <= cnt4; f += 4 * THREADS) {
            float4 v = *(const float4*)(in + gOff + f);
            *(float4*)&lds_in[f] = v;
        }
        for (int f = cnt4 + tid; f < cnt; f += THREADS) {
            lds_in[f] = in[gOff + f];
        }
    }
    __syncthreads();

    // --- Per-wave: 16 rows via WMMA (diag of A x B) ---
    const int lane = tid & 31;
    const int wv = tid >> 5;
    const int l = lane & 15;
    const int h = lane >> 4;

    const long long r0 = rowBase + (long long)wv * 16;
    const long long rowj = r0 + l;

    int m = NMAT;  // zero row
    if (rowj < Nrows) {
        long long code = tgt[rowj];
        m = (code == 255) ? NMAT : (int)code;
    }

    const float* arow = &lds_in[(wv * 16 + l) * BANDS];
    const float* brow = &lds_tab[m * BANDS];

    float dotv = 0.0f, nrmv = 0.0f;

#if HAVE_WMMA_F32
    v8f cd = {};
    v8f cn = {};
    #pragma unroll
    for (int s = 0; s < 8; ++s) {
        const int kb = 4 * s + 2 * h;
        v2f a, bt;
        a.x = arow[kb];
        a.y = (kb + 1 < BANDS) ? arow[kb + 1] : 0.0f;   // pad band 31 -> 0
        bt.x = brow[kb];
        bt.y = (kb + 1 < BANDS) ? brow[kb + 1] : 0.0f;
        cd = wmma_f32_k4(a, bt, cd);   // D[i][j] = dot(in_row_i, mat_row_j)
        cn = wmma_f32_k4(a, a, cn);    // D[i][i] = |in_row_i|^2 (B image == A image)
    }
    // Diagonal lives at: i in 0..7  -> lane i,     component i
    //                    i in 8..15 -> lane i+16,  component i-8
    {
        const bool lo = (lane < 8);
        const bool hi = (lane >= 24);
        if (lo || hi) {
            const int i = lo ? lane : (lane - 16);
            const int comp = i & 7;
            dotv = pick8(cd, comp);
            nrmv = pick8(cn, comp);
        }
    }
#else
    // Fallback: scalar dot per diagonal lane
    {
        const bool lo = (lane < 8);
        const bool hi = (lane >= 24);
        if (lo || hi) {
            const int i = lo ? lane : (lane - 16);
            const float* ar = &lds_in[(wv * 16 + i) * BANDS];
            const float* br = &lds_tab[m * BANDS];
            #pragma unroll
            for (int c = 0; c < BANDS; ++c) {
                dotv += ar[c] * br[c];
                nrmv += ar[c] * ar[c];
            }
        }
    }
#endif

    // --- Angle for the 16 diagonal lanes of this wave ---
    float ang = 0.0f;
    {
        const bool lo = (lane < 8);
        const bool hi = (lane >= 24);
        if (lo || hi) {
            const int i = lo ? lane : (lane - 16);
            const long long row = r0 + i;
            if (row < Nrows) {
                const float ntg = lds_nrm[m];   // material of this lane matches row i
                float cosv = dotv / (sqrtf(nrmv) * ntg);
                cosv = fminf(1.0f, fmaxf(-1.0f, cosv));
                ang = acosf(cosv);
            }
        }
    }

    // --- Deterministic wave reduce, then block partial ---
    #pragma unroll
    for (int off = 16; off > 0; off >>= 1)
        ang += __shfl_down(ang, off, 32);
    if (lane == 0) lds_part[wv] = ang;
    __syncthreads();
    if (tid == 0) {
        float s = 0.0f;
        #pragma unroll
        for (int w = 0; w < THREADS / 32; ++w) s += lds_part[w];
        partial[blockIdx.x] = s;
    }
}

__global__ void __launch_bounds__(THREADS)
sad_reduce_kernel(const float* __restrict__ partial, int nblocks,
                  float* __restrict__ out) {
    __shared__ float sm[THREADS];
    const int tid = threadIdx.x;
    float s = 0.0f;
    for (int i = tid; i < nblocks; i += THREADS) s += partial[i];  // fixed order per thread
    sm[tid] = s;
    __syncthreads();
    #pragma unroll
    for (int off = THREADS / 2; off > 0; off >>= 1) {
        if (tid < off) sm[tid] += sm[tid + off];
        __syncthreads();
    }
    if (tid == 0) out[0] = sm[0];
}

extern "C" void kernel_launch(void* const* d_in, const int* in_sizes, int n_in,
                              void* d_out, int out_size, void* d_ws, size_t ws_size,
                              hipStream_t stream) {
    const float* in = (const float*)d_in[0];        // [B,31,H,W] f32
    const long long* tgt = (const long long*)d_in[1];  // [B,H,W,1] i64
    const float* mat = (const float*)d_in[2];       // [16,31] f32

    const long long totalFloats = (long long)in_sizes[0];
    const long long Nrows = totalFloats / BANDS;
    const int nblocks = (int)((Nrows + ROWS_PER_BLOCK - 1) / ROWS_PER_BLOCK);

    float* partial = (float*)d_ws;  // nblocks floats (64 KB for the ref shape)

    sad_partial_kernel<<<nblocks, THREADS, 0, stream>>>(in, tgt, mat, partial, Nrows);
    sad_reduce_kernel<<<1, THREADS, 0, stream>>>(partial, nblocks, (float*)d_out);
}